// ROIHeads_2267742732667
// MI455X (gfx1250) — compile-verified
//
#include <hip/hip_runtime.h>
#include <stdint.h>

typedef uint32_t u32;
typedef uint64_t u64;
typedef unsigned int v4u __attribute__((ext_vector_type(4)));
typedef int          v4i __attribute__((ext_vector_type(4)));
typedef int          v8i __attribute__((ext_vector_type(8)));

#define Bz 16
#define Nz 8192
#define Mz 512
#define NTz (Nz + Mz)          // 8704
#define SORTN 16384            // next pow2 >= NTz
#define NCLS 80
#define NPOS 128               // 512 * 0.25
#define NBATCH 512

// ---------------------------------------------------------------------------
// CDNA5 Tensor Data Mover: DMA a contiguous 1-D tile of `nelem` elements
// (element size = 1<<dsz_code bytes) from global memory into LDS.
// D# packed per CDNA5 ISA:
//   group0 = {count=1, lds_addr, global_addr, type=2}
//   group1 = {data_size, tensor_dim0=nelem, tensor_dim1=1, tile_dim0=nelem,
//             tile_dim1=1, tensor_dim0_stride=nelem}
// Tracked with TENSORcnt; caller's wave blocks on s_wait_tensorcnt 0.
// nelem must fit in 16 bits (tile_dim0).
// ---------------------------------------------------------------------------
__device__ __forceinline__ void tdm_load_1d(const void* gsrc, u32 lds_byte_addr,
                                            u32 nelem, u32 dsz_code) {
  u64 ga = (u64)(uintptr_t)gsrc;
  v4u g0;
  g0.x = 1u;                                              // count=1, user D#
  g0.y = lds_byte_addr;                                   // lds_addr [63:32]
  g0.z = (u32)ga;                                         // global_addr[31:0]
  g0.w = (u32)((ga >> 32) & 0x01FFFFFFu) | 0x80000000u;   // addr[56:32], type=2
  v8i g1;
  g1[0] = (int)(dsz_code << 16);                          // data_size @17:16
  g1[1] = (int)((nelem & 0xFFFFu) << 16);                 // tensor_dim0 lo16 @63:48
  g1[2] = (int)((nelem >> 16) | (1u << 16));              // tensor_dim0 hi, tensor_dim1=1
  g1[3] = (int)((nelem & 0xFFFFu) << 16);                 // tile_dim0 @127:112
  g1[4] = 1;                                              // tile_dim1=1
  g1[5] = (int)nelem;                                     // tensor_dim0_stride lo32
  g1[6] = 0;
  g1[7] = 0;
  v4i z4 = {0, 0, 0, 0};
#if defined(__clang_major__) && (__clang_major__ >= 23)
  v8i z8 = {0, 0, 0, 0, 0, 0, 0, 0};
  __builtin_amdgcn_tensor_load_to_lds(g0, g1, z4, z4, z8, 0);
#else
  __builtin_amdgcn_tensor_load_to_lds(g0, g1, z4, z4, 0);
#endif
  __builtin_amdgcn_s_wait_tensorcnt(0);
}

// ---------------------------------------------------------------------------
// Kernel 1: per (batch, column) matcher.
// grid = (NT/256, B), block = 256 (8 wave32s).  gt boxes staged via TDM.
// Produces matched_idx, per-proposal class, and the 64-bit sampling key:
//   key = seg[2b] << 62 | float_bits(rand) << 30 | (16383 - j)
//   seg: 2=fg(label 1), 1=bg(label 0), 0=ignored(label -1)
// rand in [0,1) => float_bits(rand) < 2^30, so fields never overlap.
// ---------------------------------------------------------------------------
__global__ __launch_bounds__(256) void k_match(
    const float* __restrict__ prop, const unsigned char* __restrict__ pvalid,
    const float* __restrict__ gt, const int* __restrict__ gtcls,
    const unsigned char* __restrict__ gvalid, const unsigned char* __restrict__ gcrowd,
    const unsigned char* __restrict__ gdiff, const float* __restrict__ rnd,
    int* __restrict__ ws_matched, int* __restrict__ ws_cls, u64* __restrict__ ws_key) {
  __shared__ float4 sgt[Mz];          // 8 KB, filled by TDM
  __shared__ float sarea[Mz];         // 2 KB
  __shared__ unsigned char sfl[Mz];   // bit0 = valid_gt, bit1 = crowd, bit2 = difficult

  const int tid = threadIdx.x;
  const int b = blockIdx.y;
  const int j = blockIdx.x * 256 + tid;   // column in [0, NT)

  if (j < Nz) __builtin_prefetch(prop + ((size_t)b * Nz + j) * 4, 0, 0);

  if (tid < 32) {  // wave 0 issues the DMA and waits TENSORcnt==0
    tdm_load_1d(gt + (size_t)b * Mz * 4, (u32)(uintptr_t)&sgt[0],
                /*nelem=*/Mz * 4, /*dsz=4B*/ 2u);
  }
  for (int i = tid; i < Mz; i += 256) {
    unsigned char v = gvalid[(size_t)b * Mz + i];
    unsigned char c = gcrowd[(size_t)b * Mz + i];
    unsigned char d = gdiff[(size_t)b * Mz + i];
    sfl[i] = (unsigned char)(((v && !c && !d) ? 1 : 0) | (c ? 2 : 0) | (d ? 4 : 0));
  }
  __syncthreads();
  for (int i = tid; i < Mz; i += 256) {
    float4 g = sgt[i];
    sarea[i] = (g.z - g.x) * (g.w - g.y);
  }
  __syncthreads();

  float4 bx;
  int bvalid;
  if (j < Nz) {
    bx = ((const float4*)prop)[(size_t)b * Nz + j];
    bvalid = pvalid[(size_t)b * Nz + j] != 0;
  } else {
    int ig = j - Nz;
    bx = sgt[ig];
    bvalid = sfl[ig] & 1;   // appended gt validity = valid & !crowd & !difficult
  }
  const float areaB = (bx.z - bx.x) * (bx.w - bx.y);

  float best = -3.402823466e38f;
  int bidx = 0;
  float cmax = -1.0f, dmax = -1.0f;
#pragma unroll 4
  for (int i = 0; i < Mz; ++i) {
    float4 g = sgt[i];
    float lx = fmaxf(g.x, bx.x);
    float ly = fmaxf(g.y, bx.y);
    float rx = fminf(g.z, bx.z);
    float ry = fminf(g.w, bx.w);
    float w = fmaxf(rx - lx, 0.0f);
    float h = fmaxf(ry - ly, 0.0f);
    float inter = w * h;
    float uni = sarea[i] + areaB - inter;
    float iou = (uni > 0.0f) ? (inter / fmaxf(uni, 1e-12f)) : 0.0f;
    unsigned char f = sfl[i];
    float iv = (f & 1) ? iou : -1.0f;
    if (iv > best) { best = iv; bidx = i; }   // strict > == argmax first occurrence
    if (f & 2) cmax = fmaxf(cmax, iou);
    if (f & 4) dmax = fmaxf(dmax, iou);
  }

  int label = (best >= 0.5f) ? 1 : 0;
  if (label == 0 && (cmax >= 0.5f || dmax >= 0.5f)) label = -1;
  if (!bvalid) label = -1;

  int cls = (label == 1) ? gtcls[(size_t)b * Mz + bidx] : ((label == 0) ? NCLS : -1);

  float r = rnd[(size_t)b * NTz + j];
  u64 seg = (u64)(label + 1);   // 1->2, 0->1, -1->0
  u64 key = (seg << 62) | ((u64)__float_as_uint(r) << 30) | (u64)(16383 - j);

  ws_matched[(size_t)b * NTz + j] = bidx;
  ws_cls[(size_t)b * NTz + j] = cls;
  ws_key[(size_t)b * NTz + j] = key;
}

// ---------------------------------------------------------------------------
// Kernel 2: per-batch subsample.  One 1024-thread block per batch; the whole
// 16384-entry key array lives in 128 KB of LDS (CDNA5 has 320 KB per WGP).
// The 68 KB key tile is DMA'd in by the Tensor Data Mover while the other
// waves zero the pad tail.  Keys are sorted descending with a bitonic
// network; the <=512 selected entries are then re-sorted on the reference's
// coarse key (seg + rand) to reproduce jax.lax.top_k ordering exactly.
// ---------------------------------------------------------------------------
__global__ __launch_bounds__(1024) void k_sample(
    const float* __restrict__ prop, const float* __restrict__ gt,
    const u64* __restrict__ ws_key, const int* __restrict__ ws_matched,
    const int* __restrict__ ws_cls, float* __restrict__ out) {
  extern __shared__ u64 sk[];   // SORTN entries = 128 KB dynamic LDS
  __shared__ int scnt[2];       // nfg, nbg

  const int tid = threadIdx.x;
  const int b = blockIdx.x;

  if (tid < 2) scnt[tid] = 0;
  if (tid < 32) {  // wave 0: DMA the whole key tile (NT x u64) into LDS
    tdm_load_1d(ws_key + (size_t)b * NTz, (u32)(uintptr_t)&sk[0],
                /*nelem=*/NTz, /*dsz=8B*/ 3u);
  }
  for (int i = NTz + tid; i < SORTN; i += 1024) sk[i] = 0ull;  // pad tail
  __syncthreads();

  int lf = 0, lb = 0;
  for (int i = tid; i < NTz; i += 1024) {
    int seg = (int)(sk[i] >> 62);
    lf += (seg == 2);
    lb += (seg == 1);
  }
  atomicAdd(&scnt[0], lf);
  atomicAdd(&scnt[1], lb);

  // bitonic sort, descending, 16384 u64 keys in LDS
  for (int kk = 2; kk <= SORTN; kk <<= 1) {
    for (int jj = kk >> 1; jj > 0; jj >>= 1) {
      __syncthreads();
      for (int i = tid; i < SORTN; i += 1024) {
        int l = i ^ jj;
        if (l > i) {
          u64 a = sk[i], c = sk[l];
          bool sw = ((i & kk) == 0) ? (a < c) : (a > c);
          if (sw) { sk[i] = c; sk[l] = a; }
        }
      }
    }
  }
  __syncthreads();

  const int nfg = scnt[0];
  const int nbg = scnt[1];
  const int num_fg = (nfg < NPOS) ? nfg : NPOS;
  const int cap_bg = NBATCH - num_fg;
  const int num_bg = (nbg < cap_bg) ? nbg : cap_bg;
  const int num_valid = num_fg + num_bg;

  // Build the 512-entry output list: top num_fg fg, then top num_bg bg,
  // re-keyed on the coarse reference key (seg + rand) for exact top_k order.
  u64 mykey = 0;
  if (tid < NBATCH && tid < num_valid) {
    int p = (tid < num_fg) ? tid : (nfg + (tid - num_fg));
    u64 k = sk[p];
    int seg = (int)(k >> 62);
    u32 rb = (u32)((k >> 30) & 0xFFFFFFFFull);
    int j = 16383 - (int)(k & 0x3FFFull);
    float f = (float)seg + __uint_as_float(rb);     // 2+rand / 1+rand, fp32
    mykey = ((u64)__float_as_uint(f) << 32) | (u64)(16383 - j);
  }
  __syncthreads();
  if (tid < NBATCH) sk[tid] = mykey;
  __syncthreads();

  // bitonic sort of the 512 selected entries, descending
  for (int kk = 2; kk <= NBATCH; kk <<= 1) {
    for (int jj = kk >> 1; jj > 0; jj >>= 1) {
      if (tid < NBATCH) {
        int l = tid ^ jj;
        if (l > tid) {
          u64 a = sk[tid], c = sk[l];
          bool sw = ((tid & kk) == 0) ? (a < c) : (a > c);
          if (sw) { sk[tid] = c; sk[l] = a; }
        }
      }
      __syncthreads();
    }
  }

  if (tid < NBATCH) {
    u64 k = sk[tid];
    u32 fb = (u32)(k >> 32);
    bool valid = fb > 0x3F000000u;   // topv > 0.5
    float4 o0 = make_float4(0.f, 0.f, 0.f, 0.f);
    float4 o1 = make_float4(0.f, 0.f, 0.f, 0.f);
    int cls = -1;
    if (valid) {
      int j = 16383 - (int)(k & 0x3FFFull);
      const float4* bb = (j < Nz)
          ? ((const float4*)prop) + ((size_t)b * Nz + j)
          : ((const float4*)gt) + ((size_t)b * Mz + (j - Nz));
      int m = ws_matched[(size_t)b * NTz + j];
      const float4* gb = ((const float4*)gt) + ((size_t)b * Mz + m);
      o0 = *bb;
      o1 = *gb;
      cls = ws_cls[(size_t)b * NTz + j];
    }
    float4* op = (float4*)(out + ((size_t)b * NBATCH + tid) * 8);
    op[0] = o0;
    op[1] = o1;
    out[(size_t)Bz * NBATCH * 8 + (size_t)b * NBATCH + tid] = (float)cls;
  }
}

// ---------------------------------------------------------------------------
extern "C" void kernel_launch(void* const* d_in, const int* in_sizes, int n_in,
                              void* d_out, int out_size, void* d_ws, size_t ws_size,
                              hipStream_t stream) {
  const float* prop = (const float*)d_in[0];
  const unsigned char* pvalid = (const unsigned char*)d_in[1];
  const float* gt = (const float*)d_in[2];
  const int* gtcls = (const int*)d_in[3];
  const unsigned char* gvalid = (const unsigned char*)d_in[4];
  const unsigned char* gcrowd = (const unsigned char*)d_in[5];
  const unsigned char* gdiff = (const unsigned char*)d_in[6];
  const float* rnd = (const float*)d_in[7];
  float* out = (float*)d_out;

  char* ws = (char*)d_ws;
  int* ws_matched = (int*)ws;                                 // B*NT ints
  int* ws_cls = (int*)(ws + (size_t)Bz * NTz * 4);            // B*NT ints
  u64* ws_key = (u64*)(ws + (size_t)2 * Bz * NTz * 4);        // B*NT u64 (8B aligned)

  // Opt in to 128KB dynamic LDS (CDNA5 WGP has 320KB).
  (void)hipFuncSetAttribute((const void*)k_sample,
                            hipFuncAttributeMaxDynamicSharedMemorySize,
                            SORTN * (int)sizeof(u64));

  k_match<<<dim3(NTz / 256, Bz), 256, 0, stream>>>(
      prop, pvalid, gt, gtcls, gvalid, gcrowd, gdiff, rnd,
      ws_matched, ws_cls, ws_key);

  k_sample<<<dim3(Bz), 1024, SORTN * sizeof(u64), stream>>>(
      prop, gt, ws_key, ws_matched, ws_cls, out);
}